// MoGeSampledLocalLoss_15779709846258
// MI455X (gfx1250) — compile-verified
//
#include <hip/hip_runtime.h>
#include <hip/hip_bf16.h>
#include <stdint.h>

// ---------------------------------------------------------------------------
// MoGe sampled local loss, fused single-pass-over-HBM implementation.
// One wave32 per patch (K = 256 pixels -> 8 elements per lane, in registers).
// Memory-bound (~75 MB @ 23.3 TB/s => ~3.2 us floor); no matmul structure, so
// the CDNA5 path we use is async global->LDS (ASYNCcnt) + wave32 shuffles.
// ---------------------------------------------------------------------------

#define WAVE            32
#define WAVES_PER_BLOCK 8
#define BLOCK_T         (WAVE * WAVES_PER_BLOCK)   // 256 threads
#define KPIX            256                        // patch_pixel_count (setup)
#define EPL             (KPIX / WAVE)              // 8 elements per lane

#if defined(__HIP_DEVICE_COMPILE__) &&                                   \
    __has_builtin(__builtin_amdgcn_global_load_async_to_lds_b128) &&     \
    __has_builtin(__builtin_amdgcn_global_load_async_to_lds_b64)
#define USE_ASYNC_LDS 1
#else
#define USE_ASYNC_LDS 0
#endif

#if USE_ASYNC_LDS
// Exact parameter types per the clang diagnostic: pointer-to-v4i/v2i in the
// global (AS1) / LDS (AS3) address spaces, non-const.
typedef int v4i __attribute__((ext_vector_type(4)));
typedef int v2i __attribute__((ext_vector_type(2)));
typedef __attribute__((address_space(1))) v4i* g_v4i_p;
typedef __attribute__((address_space(3))) v4i* l_v4i_p;
typedef __attribute__((address_space(1))) v2i* g_v2i_p;
typedef __attribute__((address_space(3))) v2i* l_v2i_p;

__device__ __forceinline__ void async_copy_b128(const void* g, void* l) {
  __builtin_amdgcn_global_load_async_to_lds_b128((g_v4i_p)(void*)(g),
                                                 (l_v4i_p)(l), 0, 0);
}
__device__ __forceinline__ void async_copy_b64(const void* g, void* l) {
  __builtin_amdgcn_global_load_async_to_lds_b64((g_v2i_p)(void*)(g),
                                                (l_v2i_p)(l), 0, 0);
}
__device__ __forceinline__ void wait_asynccnt0() {
#if __has_builtin(__builtin_amdgcn_s_wait_asynccnt)
  __builtin_amdgcn_s_wait_asynccnt(0);
#else
  asm volatile("s_wait_asynccnt 0" ::: "memory");
#endif
  asm volatile("" ::: "memory");  // keep LDS reads after the wait
}
#endif

__device__ __forceinline__ float wave_sum(float v) {
#pragma unroll
  for (int off = WAVE / 2; off > 0; off >>= 1)
    v += __shfl_xor(v, off, WAVE);
  return v;
}

__global__ void __launch_bounds__(BLOCK_T) moge_init_kernel(float* acc) {
  if (threadIdx.x < 2) acc[threadIdx.x] = 0.0f;
}

__global__ void __launch_bounds__(BLOCK_T) moge_patch_kernel(
    const float* __restrict__ pred, const float* __restrict__ gt,
    const unsigned char* __restrict__ mask, const float* __restrict__ gscale,
    float* __restrict__ acc, int B, int P) {
  const int lane        = threadIdx.x & (WAVE - 1);
  const int wave_in_blk = threadIdx.x >> 5;
  const int wave_id     = blockIdx.x * WAVES_PER_BLOCK + wave_in_blk;
  const int nwaves      = gridDim.x * WAVES_PER_BLOCK;
  const int total       = B * P;

  __shared__ float s_acc[WAVES_PER_BLOCK * 2];
#if USE_ASYNC_LDS
  __shared__ __align__(16) float         lds_x[WAVES_PER_BLOCK][KPIX];
  __shared__ __align__(16) float         lds_y[WAVES_PER_BLOCK][KPIX];
  __shared__ __align__(16) unsigned char lds_m[WAVES_PER_BLOCK][KPIX];
#endif

  float lsum = 0.0f, lcnt = 0.0f;

  for (int p = wave_id; p < total; p += nwaves) {
    const long long base = (long long)p * KPIX;
    float x[EPL], y[EPL], m[EPL], w[EPL];

#if USE_ASYNC_LDS
    // DMA the whole 2.25KB patch tile into this wave's private LDS slab.
    {
      const char* gx = (const char*)(pred + base) + lane * 16;
      const char* gy = (const char*)(gt + base) + lane * 16;
      const char* gm = (const char*)(mask + base) + lane * 8;
      char*       lx = (char*)&lds_x[wave_in_blk][0] + lane * 16;
      char*       ly = (char*)&lds_y[wave_in_blk][0] + lane * 16;
      char*       lm = (char*)&lds_m[wave_in_blk][0] + lane * 8;
      async_copy_b128(gx, lx);
      async_copy_b128(gx + 512, lx + 512);
      async_copy_b128(gy, ly);
      async_copy_b128(gy + 512, ly + 512);
      async_copy_b64(gm, lm);
      wait_asynccnt0();
      const float4* xs = (const float4*)&lds_x[wave_in_blk][0];
      const float4* ys = (const float4*)&lds_y[wave_in_blk][0];
      float4 x0 = xs[lane * 2], x1 = xs[lane * 2 + 1];
      float4 y0 = ys[lane * 2], y1 = ys[lane * 2 + 1];
      x[0] = x0.x; x[1] = x0.y; x[2] = x0.z; x[3] = x0.w;
      x[4] = x1.x; x[5] = x1.y; x[6] = x1.z; x[7] = x1.w;
      y[0] = y0.x; y[1] = y0.y; y[2] = y0.z; y[3] = y0.w;
      y[4] = y1.x; y[5] = y1.y; y[6] = y1.z; y[7] = y1.w;
      uint2 mm = ((const uint2*)&lds_m[wave_in_blk][0])[lane];
#pragma unroll
      for (int i = 0; i < 4; ++i) m[i] = (float)((mm.x >> (8 * i)) & 0xffu);
#pragma unroll
      for (int i = 0; i < 4; ++i) m[4 + i] = (float)((mm.y >> (8 * i)) & 0xffu);
    }
#else
    // Direct b128 register loads (coalesced); prefetch hints on gfx1250 lower
    // to global_prefetch_b8.
    {
      __builtin_prefetch(gt + base + lane * 4, 0, 1);
      __builtin_prefetch(mask + base + lane, 0, 1);
      const float4* xs = (const float4*)(pred + base);
      const float4* ys = (const float4*)(gt + base);
      float4 x0 = xs[lane * 2], x1 = xs[lane * 2 + 1];
      float4 y0 = ys[lane * 2], y1 = ys[lane * 2 + 1];
      x[0] = x0.x; x[1] = x0.y; x[2] = x0.z; x[3] = x0.w;
      x[4] = x1.x; x[5] = x1.y; x[6] = x1.z; x[7] = x1.w;
      y[0] = y0.x; y[1] = y0.y; y[2] = y0.z; y[3] = y0.w;
      y[4] = y1.x; y[5] = y1.y; y[6] = y1.z; y[7] = y1.w;
      uint2 mm = ((const uint2*)(mask + base))[lane];
#pragma unroll
      for (int i = 0; i < 4; ++i) m[i] = (float)((mm.x >> (8 * i)) & 0xffu);
#pragma unroll
      for (int i = 0; i < 4; ++i) m[4 + i] = (float)((mm.y >> (8 * i)) & 0xffu);
    }
#endif

    // ---- pass 1: mask stats + WLS sums -----------------------------------
    float p_m = 0.f, p_my = 0.f;
    float p_w = 0.f, p_wx = 0.f, p_wy = 0.f, p_wxx = 0.f, p_wxy = 0.f;
#pragma unroll
    for (int i = 0; i < EPL; ++i) {
      float mi = m[i], xi = x[i], yi = y[i];
      float wi = mi / fmaxf(yi, 1e-5f);
      w[i] = wi;
      p_m += mi;
      p_my += mi * yi;
      p_w += wi;
      p_wx += wi * xi;
      p_wy += wi * yi;
      p_wxx += wi * xi * xi;
      p_wxy += wi * xi * yi;
    }
    float msum = wave_sum(p_m);
    float symsum = wave_sum(p_my);
    float sw = wave_sum(p_w);
    float swx = wave_sum(p_wx);
    float swy = wave_sum(p_wy);
    float swxx = wave_sum(p_wxx);
    float swxy = wave_sum(p_wxy);

    float det = sw * swxx - swx * swx;
    det = (fabsf(det) < 1e-12f) ? 1e-12f : det;
    float s0 = (sw * swxy - swx * swy) / det;
    float t0 = (swxx * swy - swx * swxy) / det;

    // ---- pass 2: truncated refit -----------------------------------------
    float q_w = 0.f, q_wx = 0.f, q_wy = 0.f, q_wxx = 0.f, q_wxy = 0.f;
#pragma unroll
    for (int i = 0; i < EPL; ++i) {
      float xi = x[i], yi = y[i];
      float e = w[i] * fabsf(s0 * xi + t0 - yi);
      float w2 = (e < 1.0f) ? w[i] : 0.0f;  // TRUNC = 1.0
      q_w += w2;
      q_wx += w2 * xi;
      q_wy += w2 * yi;
      q_wxx += w2 * xi * xi;
      q_wxy += w2 * xi * yi;
    }
    float rw = wave_sum(q_w);
    float rwx = wave_sum(q_wx);
    float rwy = wave_sum(q_wy);
    float rwxx = wave_sum(q_wxx);
    float rwxy = wave_sum(q_wxy);

    // Fallback w2 -> w is identical to reusing the pass-1 sums.
    bool use2 = (rw > 1e-8f);
    float SW = use2 ? rw : sw;
    float SWX = use2 ? rwx : swx;
    float SWY = use2 ? rwy : swy;
    float SWXX = use2 ? rwxx : swxx;
    float SWXY = use2 ? rwxy : swxy;

    float det2 = SW * SWXX - SWX * SWX;
    det2 = (fabsf(det2) < 1e-12f) ? 1e-12f : det2;
    float s = (SW * SWXY - SWX * SWY) / det2;
    float t = (SWXX * SWY - SWX * SWXY) / det2;

    // ---- validity ---------------------------------------------------------
    float valid_ratio = msum * (1.0f / (float)KPIX);
    float mean_depth = (msum > 0.0f) ? (symsum / fmaxf(msum, 1.0f)) : 1.0f;
    float gs = gscale[p / P];
    float ratio = s / fmaxf(gs, 1e-12f);
    bool gs_ok = (gs > 0.0f) ? (ratio >= 0.1f && ratio <= 10.0f) : true;
    bool valid = (valid_ratio >= 0.3f) && (s > 0.0f) && gs_ok;

    // ---- pass 3: smooth-L1 patch loss ------------------------------------
    float dfloor = 0.1f * mean_depth;
    float p_l = 0.f;
#pragma unroll
    for (int i = 0; i < EPL; ++i) {
      float yi = y[i];
      float pw = m[i] / fmaxf(yi, dfloor);
      float r = fabsf(s * x[i] + t - yi) * pw;
      // beta = 0.1: 0.5*r*r/beta = 5*r*r ; r - 0.5*beta = r - 0.05
      p_l += (r < 0.1f) ? (5.0f * r * r) : (r - 0.05f);
    }
    float patch_loss = wave_sum(p_l) * (1.0f / (float)KPIX);

    if (valid) {  // wave-uniform predicate
      lsum += patch_loss;
      lcnt += 1.0f;
    }
  }

  // ---- block reduction + global accumulation -----------------------------
  if (lane == 0) {
    s_acc[wave_in_blk * 2 + 0] = lsum;
    s_acc[wave_in_blk * 2 + 1] = lcnt;
  }
  __syncthreads();
  if (threadIdx.x == 0) {
    float a = 0.f, c = 0.f;
#pragma unroll
    for (int i = 0; i < WAVES_PER_BLOCK; ++i) {
      a += s_acc[2 * i + 0];
      c += s_acc[2 * i + 1];
    }
    atomicAdd(&acc[0], a);
    atomicAdd(&acc[1], c);
  }
}

__global__ void moge_final_kernel(const float* __restrict__ acc,
                                  float* __restrict__ out) {
  if (threadIdx.x == 0 && blockIdx.x == 0) {
    float c = acc[1];
    out[0] = (c > 0.0f) ? (acc[0] / fmaxf(c, 1.0f)) : 0.0f;
  }
}

extern "C" void kernel_launch(void* const* d_in, const int* in_sizes, int n_in,
                              void* d_out, int out_size, void* d_ws,
                              size_t ws_size, hipStream_t stream) {
  (void)n_in; (void)out_size; (void)ws_size;
  const float*         pred  = (const float*)d_in[0];
  const float*         gt    = (const float*)d_in[1];
  const unsigned char* mask  = (const unsigned char*)d_in[2];  // bool array
  const float*         gs    = (const float*)d_in[3];
  // d_in[4]/d_in[5] are the scalar ints num_patches / patch_pixel_count on
  // device; K is fixed by the setup (256) and required at compile time for
  // the register tiling, so derive P on the host from the flat sizes.
  int       B = in_sizes[3];
  long long N = (long long)in_sizes[0] / (B > 0 ? B : 1);
  int       P = (int)(N / KPIX);

  float* acc = (float*)d_ws;  // [0]=sum(patch_loss*valid), [1]=count(valid)

  moge_init_kernel<<<1, BLOCK_T, 0, stream>>>(acc);

  long long total  = (long long)B * P;
  int       blocks = (int)((total + WAVES_PER_BLOCK - 1) / WAVES_PER_BLOCK);
  if (blocks < 1) blocks = 1;
  moge_patch_kernel<<<blocks, BLOCK_T, 0, stream>>>(pred, gt, mask, gs, acc, B, P);

  moge_final_kernel<<<1, 32, 0, stream>>>(acc, (float*)d_out);
}